// GaussianProcessCholeskyAdvanced_7026566496647
// MI455X (gfx1250) — compile-verified
//
#include <hip/hip_runtime.h>
#include <math.h>

// ---------------------------------------------------------------------------
// GP regression with Cholesky for MI455X (gfx1250, wave32, WMMA).
// Dominant cost: ~850 GFLOP of f32 GEMM work (trsm updates + cov = Kss - v^T v)
// -> routed through V_WMMA_F32_16X16X4_F32. Memory floor (cov write 268MB)
// is ~12us at 23.3 TB/s, so compute-bound; f32 WMMA keeps exact f32 FMA
// semantics for the numerically sensitive Cholesky/solve chain.
// LDS tiles are stored as k-PAIRED float2 so each WMMA A/B fragment is a
// single aligned ds_load_b64 straight into the operand register pair.
// ---------------------------------------------------------------------------

#define N_TRAIN 4096
#define N_TEST  8192
#define DIMS    8
#define NB      128      // Cholesky / trsm panel width
#define TP_ROWS 64       // rows per trsm_panel workgroup
#define TV_COLS 64       // columns per trsm_diag_v workgroup

#define BM 128
#define BN 128
#define BK 32
#define GEMM_THREADS 256 // 8 wave32s: 4x2 wave grid, wave tile 32x64

typedef float v2f __attribute__((ext_vector_type(2)));
typedef float v8f __attribute__((ext_vector_type(8)));

enum { EPI_SUB = 0, EPI_COV = 1 };

__device__ __forceinline__ float softplus_f(float x) {
  return (x > 20.f) ? x : log1pf(expf(x));
}

// p[0] = 0.5/ls^2, p[1] = var, p[2] = softplus(raw_noise) + 1e-6
__global__ void params_kernel(const float* rls, const float* rvar,
                              const float* rnz, float* p) {
  if (threadIdx.x == 0) {
    float ls  = softplus_f(rls[0]);
    float var = softplus_f(rvar[0]);
    float nz  = softplus_f(rnz[0]);
    p[0] = 0.5f / (ls * ls);
    p[1] = var;
    p[2] = nz + 1e-6f;
  }
}

// weighted_K: diag = k + noise_tot ; off-diag = k * w_i * w_j
__global__ void rbf_train_kernel(const float* __restrict__ x,
                                 const float* __restrict__ w,
                                 const float* __restrict__ p,
                                 float* __restrict__ K, int n) {
  __shared__ float Xi[16][DIMS];
  __shared__ float Xj[16][DIMS];
  const int tx = threadIdx.x, ty = threadIdx.y;
  const int t = ty * 16 + tx;
  const size_t i0 = (size_t)blockIdx.y * 16, j0 = (size_t)blockIdx.x * 16;
  if (t < 16 * DIMS) {
    int r = t / DIMS, d = t % DIMS;
    Xi[r][d] = x[(i0 + r) * DIMS + d];
    Xj[r][d] = x[(j0 + r) * DIMS + d];
  }
  __syncthreads();
  float d2 = 0.f;
#pragma unroll
  for (int d = 0; d < DIMS; d++) {
    float df = Xi[ty][d] - Xj[tx][d];
    d2 += df * df;
  }
  const size_t i = i0 + ty, j = j0 + tx;
  float k = p[1] * expf(-d2 * p[0]);
  float val = (i == j) ? (k + p[2]) : (k * w[i] * w[j]);
  K[i * (size_t)n + j] = val;
}

// V[i,j] = w[i] * var * exp(-0.5 d2(xtr_i, xte_j)/ls^2)
__global__ void rbf_cross_kernel(const float* __restrict__ xtr,
                                 const float* __restrict__ xte,
                                 const float* __restrict__ w,
                                 const float* __restrict__ p,
                                 float* __restrict__ V, int n, int m) {
  __shared__ float Xi[16][DIMS];
  __shared__ float Xj[16][DIMS];
  const int tx = threadIdx.x, ty = threadIdx.y;
  const int t = ty * 16 + tx;
  const size_t i0 = (size_t)blockIdx.y * 16, j0 = (size_t)blockIdx.x * 16;
  if (t < 16 * DIMS) {
    int r = t / DIMS, d = t % DIMS;
    Xi[r][d] = xtr[(i0 + r) * DIMS + d];
    Xj[r][d] = xte[(j0 + r) * DIMS + d];
  }
  __syncthreads();
  float d2 = 0.f;
#pragma unroll
  for (int d = 0; d < DIMS; d++) {
    float df = Xi[ty][d] - Xj[tx][d];
    d2 += df * df;
  }
  const size_t i = i0 + ty, j = j0 + tx;
  V[i * (size_t)m + j] = w[i] * p[1] * expf(-d2 * p[0]);
}

// ---------------------------------------------------------------------------
// Generic WMMA-f32 GEMM tile kernel.
//   acc[m,n] = sum_k A(m,k) * B(k,n)    with A(m,k) = TA ? A[k*lda+m] : A[m*lda+k]
//                                            B(k,n) = TB ? B[n*ldb+k] : B[k*ldb+n]
//   EPI_SUB: C[m,n] -= acc
//   EPI_COV: C[i,j] = C[j,i] = Kss(i,j) - acc   (lower-triangle blocks only)
// LDS tiles hold k-pairs: As2[kp][m] = { A(m, 2kp), A(m, 2kp+1) }, so the
// 16x4 f32 A fragment (lanes 0-15: K={0,1}, lanes 16-31: K={2,3}) and the
// 4x16 B fragment are each one aligned ds_load_b64.
// ---------------------------------------------------------------------------
template <bool TA, bool TB, int EPI>
__launch_bounds__(GEMM_THREADS)
__global__ void gemm_kernel(const float* __restrict__ A, int lda,
                            const float* __restrict__ B, int ldb,
                            float* __restrict__ C, int ldc, int Ktot,
                            const float* __restrict__ xt,
                            const float* __restrict__ p) {
  __shared__ v2f As2[BK / 2][BM];
  __shared__ v2f Bs2[BK / 2][BN];
  __shared__ float Xi[BM][DIMS];
  __shared__ float Xj[BN][DIMS];

  const int tid = threadIdx.x;
  const int lane = tid & 31;
  const int half = lane >> 4;    // 0/1 : selects K-pair / row half
  const int l15 = lane & 15;
  const int waveId = tid >> 5;
  const int wm = (waveId & 3) * 32;  // wave row offset in block tile
  const int wn = (waveId >> 2) * 64; // wave col offset in block tile

  const int bx = blockIdx.x, by = blockIdx.y;
  if (EPI == EPI_COV && bx > by) return;  // symmetry: lower blocks only
  const size_t m0 = (size_t)by * BM;
  const size_t n0 = (size_t)bx * BN;

  if (EPI == EPI_COV) {
    for (int idx = tid; idx < (BM * DIMS) / 4; idx += GEMM_THREADS) {
      int r = idx >> 1, part = (idx & 1) * 4;
      *(float4*)&Xi[r][part] = *(const float4*)&xt[(m0 + r) * DIMS + part];
      *(float4*)&Xj[r][part] = *(const float4*)&xt[(n0 + r) * DIMS + part];
    }
  }

  v8f acc[2][4] = {};

  for (int k0 = 0; k0 < Ktot; k0 += BK) {
    // ---- stage A tile into LDS as k-paired As2[kp][m] ----
    if (TA) {
      // global rows are k (contiguous in m): load 2 k-rows, interleave
      for (int idx = tid; idx < (BK / 2) * (BM / 2); idx += GEMM_THREADS) {
        int kp = idx / (BM / 2);
        int mp = (idx % (BM / 2)) * 2;
        float2 r0 = *(const float2*)&A[(size_t)(k0 + 2 * kp) * lda + m0 + mp];
        float2 r1 =
            *(const float2*)&A[(size_t)(k0 + 2 * kp + 1) * lda + m0 + mp];
        v2f e0; e0.x = r0.x; e0.y = r1.x;
        v2f e1; e1.x = r0.y; e1.y = r1.y;
        As2[kp][mp] = e0;
        As2[kp][mp + 1] = e1;
      }
    } else {
      // global rows are m (contiguous in k): float4 holds 2 k-pairs
      for (int idx = tid; idx < BM * BK / 4; idx += GEMM_THREADS) {
        int mm = idx / (BK / 4), kq = (idx % (BK / 4)) * 4;
        float4 t = *(const float4*)&A[(m0 + mm) * (size_t)lda + k0 + kq];
        v2f lo; lo.x = t.x; lo.y = t.y;
        v2f hv; hv.x = t.z; hv.y = t.w;
        As2[(kq >> 1)][mm] = lo;
        As2[(kq >> 1) + 1][mm] = hv;
      }
    }
    // ---- stage B tile into LDS as k-paired Bs2[kp][n] ----
    if (TB) {
      for (int idx = tid; idx < BN * BK / 4; idx += GEMM_THREADS) {
        int nn = idx / (BK / 4), kq = (idx % (BK / 4)) * 4;
        float4 t = *(const float4*)&B[(n0 + nn) * (size_t)ldb + k0 + kq];
        v2f lo; lo.x = t.x; lo.y = t.y;
        v2f hv; hv.x = t.z; hv.y = t.w;
        Bs2[(kq >> 1)][nn] = lo;
        Bs2[(kq >> 1) + 1][nn] = hv;
      }
    } else {
      for (int idx = tid; idx < (BK / 2) * (BN / 2); idx += GEMM_THREADS) {
        int kp = idx / (BN / 2);
        int np = (idx % (BN / 2)) * 2;
        float2 r0 = *(const float2*)&B[(size_t)(k0 + 2 * kp) * ldb + n0 + np];
        float2 r1 =
            *(const float2*)&B[(size_t)(k0 + 2 * kp + 1) * ldb + n0 + np];
        v2f e0; e0.x = r0.x; e0.y = r1.x;
        v2f e1; e1.x = r0.y; e1.y = r1.y;
        Bs2[kp][np] = e0;
        Bs2[kp][np + 1] = e1;
      }
    }
    __syncthreads();

    if (k0 + BK < Ktot) {  // hint next tiles into cache (global_prefetch_b8)
      const float* pa = TA ? &A[(size_t)(k0 + BK) * lda + m0 + tid]
                           : &A[(m0 + (tid >> 1)) * (size_t)lda + k0 + BK];
      const float* pb = TB ? &B[(n0 + (tid >> 1)) * (size_t)ldb + k0 + BK]
                           : &B[(size_t)(k0 + BK) * ldb + n0 + tid];
      __builtin_prefetch(pa, 0, 1);
      __builtin_prefetch(pb, 0, 1);
    }

#pragma unroll
    for (int kk = 0; kk < BK; kk += 4) {
      const int kp = (kk >> 1) + half;  // (kk + 2*half) / 2
      v2f a[2];
      v2f b[4];
#pragma unroll
      for (int fm = 0; fm < 2; fm++) a[fm] = As2[kp][wm + fm * 16 + l15];
#pragma unroll
      for (int fn = 0; fn < 4; fn++) b[fn] = Bs2[kp][wn + fn * 16 + l15];
#pragma unroll
      for (int fm = 0; fm < 2; fm++)
#pragma unroll
        for (int fn = 0; fn < 4; fn++)
          acc[fm][fn] = __builtin_amdgcn_wmma_f32_16x16x4_f32(
              false, a[fm], false, b[fn], (short)0, acc[fm][fn], false, false);
    }
    __syncthreads();
  }

  // ---- epilogue ----
  const float inv2 = (EPI == EPI_COV) ? p[0] : 0.f;
  const float var  = (EPI == EPI_COV) ? p[1] : 0.f;
#pragma unroll
  for (int fm = 0; fm < 2; fm++)
#pragma unroll
    for (int fn = 0; fn < 4; fn++)
#pragma unroll
      for (int r = 0; r < 8; r++) {
        const float vacc = acc[fm][fn][r];
        const int lm = wm + fm * 16 + half * 8 + r;
        const int ln = wn + fn * 16 + l15;
        const size_t gi = m0 + lm, gj = n0 + ln;
        if (EPI == EPI_SUB) {
          C[gi * (size_t)ldc + gj] -= vacc;
        } else {
          float d2 = 0.f;
#pragma unroll
          for (int d = 0; d < DIMS; d++) {
            float df = Xi[lm][d] - Xj[ln][d];
            d2 += df * df;
          }
          float kss = var * expf(-d2 * inv2);
          if (gi == gj) kss += 1e-6f;
          const float outv = kss - vacc;
          C[gi * (size_t)ldc + gj] = outv;
          C[gj * (size_t)ldc + gi] = outv;  // symmetric mirror
        }
      }
}

// ---- unblocked Cholesky of the NB x NB diagonal block (1 WG, LDS) ----
__global__ void potrf_diag_kernel(float* __restrict__ Kw, int n, int k0) {
  extern __shared__ float smem[];
  float(*Ls)[NB] = (float(*)[NB])smem;
  const int tid = threadIdx.x;
  for (int idx = tid; idx < NB * NB; idx += blockDim.x)
    Ls[idx / NB][idx % NB] = Kw[(size_t)(k0 + idx / NB) * n + k0 + idx % NB];
  __syncthreads();
  for (int j = 0; j < NB; j++) {
    if (tid == 0) Ls[j][j] = sqrtf(Ls[j][j]);
    __syncthreads();
    const float dinv = 1.0f / Ls[j][j];
    for (int i = j + 1 + tid; i < NB; i += blockDim.x) Ls[i][j] *= dinv;
    __syncthreads();
    const int rem = NB - j - 1;
    for (int idx = tid; idx < rem * rem; idx += blockDim.x) {
      int i = j + 1 + idx / rem, t = j + 1 + idx % rem;
      Ls[i][t] -= Ls[i][j] * Ls[t][j];
    }
    __syncthreads();
  }
  for (int idx = tid; idx < NB * NB; idx += blockDim.x)
    Kw[(size_t)(k0 + idx / NB) * n + k0 + idx % NB] = Ls[idx / NB][idx % NB];
}

// ---- panel solve: X * L_kk^T = B, one row per thread, per-row slice in LDS ----
__global__ void trsm_panel_kernel(float* __restrict__ Kw, int n, int k0,
                                  int row0) {
  extern __shared__ float smem[];
  float(*Ls)[NB] = (float(*)[NB])smem;                       // 128x128
  float(*Xs)[NB + 1] = (float(*)[NB + 1])(smem + NB * NB);   // 64x129 (pad)
  const int tid = threadIdx.x;  // 0..TP_ROWS-1
  for (int idx = tid; idx < NB * NB; idx += TP_ROWS)
    Ls[idx / NB][idx % NB] = Kw[(size_t)(k0 + idx / NB) * n + k0 + idx % NB];
  const size_t grow = (size_t)row0 + (size_t)blockIdx.x * TP_ROWS + tid;
  for (int j = 0; j < NB; j++) Xs[tid][j] = Kw[grow * n + k0 + j];
  __syncthreads();
  for (int j = 0; j < NB; j++) {
    const float x = Xs[tid][j] / Ls[j][j];
    Xs[tid][j] = x;
    for (int u = j + 1; u < NB; u++) Xs[tid][u] -= Ls[u][j] * x;
  }
  for (int j = 0; j < NB; j++) Kw[grow * n + k0 + j] = Xs[tid][j];
}

// ---- diag solve for V: L_kk * X = V_block, one column per thread ----
__global__ void trsm_diag_v_kernel(const float* __restrict__ Kw,
                                   float* __restrict__ V, int n, int m,
                                   int k0) {
  extern __shared__ float smem[];
  float(*Ls)[NB] = (float(*)[NB])smem;                          // 128x128
  float(*Xs)[TV_COLS] = (float(*)[TV_COLS])(smem + NB * NB);    // 128x64
  const int tid = threadIdx.x;  // 0..TV_COLS-1
  const size_t col = (size_t)blockIdx.x * TV_COLS + tid;
  for (int idx = tid; idx < NB * NB; idx += TV_COLS)
    Ls[idx / NB][idx % NB] = Kw[(size_t)(k0 + idx / NB) * n + k0 + idx % NB];
  for (int r = 0; r < NB; r++) Xs[r][tid] = V[(size_t)(k0 + r) * m + col];
  __syncthreads();
  for (int t = 0; t < NB; t++) {
    const float x = Xs[t][tid] / Ls[t][t];
    Xs[t][tid] = x;
    for (int r = t + 1; r < NB; r++) Xs[r][tid] -= Ls[r][t] * x;
  }
  for (int r = 0; r < NB; r++) V[(size_t)(k0 + r) * m + col] = Xs[r][tid];
}

// ---- cho_solve for alpha: L z = y, L^T a = z (single WG, z/a in LDS) ----
__global__ void solve_alpha_kernel(const float* __restrict__ L,
                                   const float* __restrict__ y,
                                   float* __restrict__ alpha_out, int n) {
  __shared__ float zs[N_TRAIN];
  __shared__ float as[N_TRAIN];
  __shared__ float red[256];
  const int tid = threadIdx.x;
  for (int i = 0; i < n; i++) {
    float part = 0.f;
    for (int t = tid; t < i; t += 256) part += L[(size_t)i * n + t] * zs[t];
    red[tid] = part;
    __syncthreads();
    for (int s = 128; s > 0; s >>= 1) {
      if (tid < s) red[tid] += red[tid + s];
      __syncthreads();
    }
    if (tid == 0) zs[i] = (y[i] - red[0]) / L[(size_t)i * n + i];
    __syncthreads();
  }
  for (int i = n - 1; i >= 0; i--) {
    float part = 0.f;
    for (int t = i + 1 + tid; t < n; t += 256)
      part += L[(size_t)t * n + i] * as[t];
    red[tid] = part;
    __syncthreads();
    for (int s = 128; s > 0; s >>= 1) {
      if (tid < s) red[tid] += red[tid + s];
      __syncthreads();
    }
    if (tid == 0) as[i] = (zs[i] - red[0]) / L[(size_t)i * n + i];
    __syncthreads();
  }
  for (int i = tid; i < n; i += 256) alpha_out[i] = as[i];
}

// ---- mu[j] = sum_i V[i,j] * alpha[i]  (V still = weighted_K_s here) ----
__global__ void mu_kernel(const float* __restrict__ V,
                          const float* __restrict__ alpha,
                          float* __restrict__ mu, int n, int m) {
  const int j = blockIdx.x * blockDim.x + threadIdx.x;
  float s = 0.f;
  for (int i = 0; i < n; i++) s += V[(size_t)i * m + j] * alpha[i];
  mu[j] = s;
}

extern "C" void kernel_launch(void* const* d_in, const int* in_sizes, int n_in,
                              void* d_out, int out_size, void* d_ws,
                              size_t ws_size, hipStream_t stream) {
  const float* x_train = (const float*)d_in[0];
  const float* y_train = (const float*)d_in[1];
  const float* w_train = (const float*)d_in[2];
  const float* x_test  = (const float*)d_in[3];
  const float* rls  = (const float*)d_in[4];
  const float* rvar = (const float*)d_in[5];
  const float* rnz  = (const float*)d_in[6];

  const int n = N_TRAIN, m = N_TEST;

  // workspace carve (~200 MB): params | K/L (64MB) | V (128MB) | alpha
  float* p  = (float*)d_ws;
  float* Kw = p + 16;
  float* V  = Kw + (size_t)n * n;
  float* al = V + (size_t)n * m;

  float* mu  = (float*)d_out;
  float* cov = mu + m;

  params_kernel<<<1, 32, 0, stream>>>(rls, rvar, rnz, p);
  rbf_train_kernel<<<dim3(n / 16, n / 16), dim3(16, 16), 0, stream>>>(
      x_train, w_train, p, Kw, n);

  // blocked right-looking Cholesky, trailing update via WMMA-f32 SYRK
  for (int k0 = 0; k0 < n; k0 += NB) {
    potrf_diag_kernel<<<1, 256, NB * NB * sizeof(float), stream>>>(Kw, n, k0);
    const int rem = n - k0 - NB;
    if (rem > 0) {
      const size_t shp = (size_t)(NB * NB + TP_ROWS * (NB + 1)) * sizeof(float);
      trsm_panel_kernel<<<rem / TP_ROWS, TP_ROWS, shp, stream>>>(Kw, n, k0,
                                                                 k0 + NB);
      gemm_kernel<false, true, EPI_SUB>
          <<<dim3(rem / BN, rem / BM), GEMM_THREADS, 0, stream>>>(
              Kw + (size_t)(k0 + NB) * n + k0, n,
              Kw + (size_t)(k0 + NB) * n + k0, n,
              Kw + (size_t)(k0 + NB) * n + (k0 + NB), n, NB, nullptr, nullptr);
    }
  }

  solve_alpha_kernel<<<1, 256, 0, stream>>>(Kw, y_train, al, n);

  rbf_cross_kernel<<<dim3(m / 16, n / 16), dim3(16, 16), 0, stream>>>(
      x_train, x_test, w_train, p, V, n, m);
  mu_kernel<<<m / 256, 256, 0, stream>>>(V, al, mu, n, m);

  // blocked forward solve L * v = weighted_K_s (in place in V)
  for (int k0 = 0; k0 < n; k0 += NB) {
    const size_t shv = (size_t)(NB * NB + NB * TV_COLS) * sizeof(float);
    trsm_diag_v_kernel<<<m / TV_COLS, TV_COLS, shv, stream>>>(Kw, V, n, m, k0);
    const int rem = n - k0 - NB;
    if (rem > 0) {
      gemm_kernel<false, false, EPI_SUB>
          <<<dim3(m / BN, rem / BM), GEMM_THREADS, 0, stream>>>(
              Kw + (size_t)(k0 + NB) * n + k0, n, V + (size_t)k0 * m, m,
              V + (size_t)(k0 + NB) * m, m, NB, nullptr, nullptr);
    }
  }

  // cov = Kss - v^T v, Kss fused in epilogue, symmetric (lower blocks only)
  gemm_kernel<true, false, EPI_COV>
      <<<dim3(m / BN, m / BM), GEMM_THREADS, 0, stream>>>(V, m, V, m, cov, m,
                                                          n, x_test, p);
}